// BatchTangentProject_15324443312270
// MI455X (gfx1250) — compile-verified
//
#include <hip/hip_runtime.h>

// ---------------------------------------------------------------------------
// BatchTangentProject for MI455X (gfx1250), GEMM-only formulation.
//
// eigh-free algorithm (all 512x512 fp32 GEMMs -> V_WMMA_F32_16X16X4_F32):
//   mean       = mean_b(input)                       (elementwise)
//   new_weight = 0.5*w + 0.5*mean                    (-> d_out tail)
//   Xr         = 0.99*sym(new_weight) + 0.01*I
//   sri        = Xr^{-1/2}  via coupled Newton-Schulz (Frobenius-normalized)
//   Q          = sri @ input_b @ sri                 (2 GEMMs per (b,c))
//   M          = 0.99*sym(Q) + 0.01*I
//   log(M)     = 2^6 * log(M^{1/64});  6 NS square roots + 8-term Mercator
//               series in R = M^{1/64} - I (Horner, GEMM per term)
//
// ~8 TFLOP of fp32 GEMM vs ~0.2 GB HBM traffic -> compute bound -> WMMA,
// with tiles staged into LDS by the Tensor Data Mover (TENSOR_LOAD_TO_LDS,
// TENSORcnt double-buffering) so DMA overlaps the matrix pipe.
// ---------------------------------------------------------------------------

typedef float v2f __attribute__((ext_vector_type(2)));
typedef float v8f __attribute__((ext_vector_type(8)));

typedef unsigned int u32x4 __attribute__((ext_vector_type(4)));
typedef int i32x4 __attribute__((ext_vector_type(4)));
typedef int i32x8 __attribute__((ext_vector_type(8)));

#define NMAT  512
#define NN    (NMAT * NMAT)
#define BATCH 32
#define CCH   4
#define PSI   0.01f

#define NS_IT_W 20   // Newton-Schulz iters for weight inverse-sqrt
#define NS_IT_L 12   // Newton-Schulz iters per log-stage sqrt
#define K_SQRT  6    // number of square roots before the log series
#define SER_M   8    // Mercator series terms

// LDS tile pitches (match TDM pad_interval/pad_amount below)
#define LDA_P 17     // A tile row pitch: 16 k + 1 pad DWORD
#define LDB_P 68     // B tile row pitch: 64 n + 4 pad DWORDs

#if defined(__has_builtin)
#if __has_builtin(__builtin_amdgcn_tensor_load_to_lds)
#define USE_TDM 1
#endif
#endif
#ifndef USE_TDM
#define USE_TDM 0
#endif

#if USE_TDM
// Issue one 2D TENSOR_LOAD_TO_LDS: tile (tile_x x tile_y) of a 512x512 fp32
// row-major tensor starting at gaddr, row stride 512, written to LDS at
// lds_off with pad_amount(+1 code) DWORDs inserted every pad_interval code'd
// DWORDs (D# group1 fields per CDNA5 ISA ch.8).
__device__ __forceinline__ void tdm_load_2d(unsigned lds_off, const float* gaddr,
                                            unsigned tile_x, unsigned tile_y,
                                            unsigned pad_interval, unsigned pad_amount)
{
  const unsigned long long ga = (unsigned long long)(uintptr_t)gaddr;
  u32x4 g0;
  g0[0] = 1u;                                              // count=1, load, no gather
  g0[1] = lds_off;                                         // lds_addr
  g0[2] = (unsigned)(ga & 0xFFFFFFFFu);                    // global_addr[95:64]
  g0[3] = (unsigned)((ga >> 32) & 0x01FFFFFFu) | (2u << 30); // addr[56:32] | type=2
  i32x8 g1;
  g1[0] = (int)((2u << 16)                                 // data_size = 4B
                | (1u << 20)                               // pad_enable
                | (pad_interval << 22) | (pad_amount << 25));
  g1[1] = (int)(512u << 16);                               // tensor_dim0 = 512
  g1[2] = (int)(512u << 16);                               // tensor_dim1 = 512
  g1[3] = (int)(tile_x << 16);                             // tile_dim0
  g1[4] = (int)tile_y;                                     // tile_dim1 (tile_dim2=0)
  g1[5] = (int)NMAT;                                       // tensor_dim0_stride = 512
  g1[6] = 0;
  g1[7] = 0;                                               // dim1_stride unused (2D)
  const i32x4 z4 = {0, 0, 0, 0};
#if __clang_major__ >= 23
  const i32x8 z8 = {0, 0, 0, 0, 0, 0, 0, 0};
  __builtin_amdgcn_tensor_load_to_lds(g0, g1, z4, z4, z8, 0);
#else
  __builtin_amdgcn_tensor_load_to_lds(g0, g1, z4, z4, 0);
#endif
}
#endif

// ---------------------------------------------------------------------------
// Batched 512x512x512 fp32 GEMM:  D = alpha * (A @ B) + delta * I
// Operand base selection per batch slice z (g = g0 + z):
//   mode 0: ptr + z*NN   mode 1: ptr + (g % CCH)*NN   mode 2: ptr + (g/CCH)*NN
// D: dMode 0: z*NN ; dMode 1: g*NN (final write into d_out)
// Block: 128 threads = 4 wave32; tile 64x64; each wave 32x32 (2x2 WMMA frags).
// Tiles double-buffered in LDS, filled by the TDM (TENSORcnt-pipelined).
// ---------------------------------------------------------------------------
__global__ __launch_bounds__(128) void gemm512_wmma(
    const float* __restrict__ A, int aMode,
    const float* __restrict__ Bm, int bMode,
    float* __restrict__ D, int dMode,
    int g0, float alpha, float delta)
{
  const int z = blockIdx.z;
  const int g = g0 + z;
  const float* Ab = A  + (size_t)((aMode == 0) ? z : (aMode == 1) ? (g % CCH) : (g / CCH)) * NN;
  const float* Bb = Bm + (size_t)((bMode == 0) ? z : (bMode == 1) ? (g % CCH) : (g / CCH)) * NN;
  float*       Db = D  + (size_t)((dMode == 0) ? z : g) * NN;

  const int tileM = blockIdx.y * 64;
  const int tileN = blockIdx.x * 64;
  const int tid   = threadIdx.x;
  const int lane  = tid & 31;
  const int wave  = tid >> 5;
  const int wm    = (wave >> 1) * 32;   // wave row base inside tile
  const int wn    = (wave & 1) * 32;    // wave col base inside tile

  __shared__ float lsA[2][64 * LDA_P];  // A tile, row-major [m][k], padded
  __shared__ float lsB[2][16 * LDB_P];  // B tile, row-major [k][n], padded

  v8f acc[2][2] = {};

  const int mr = lane & 15;             // fragment row/col within 16
  const int kb = (lane >> 4) << 1;      // K sub: lanes 0-15 -> {0,1}, 16-31 -> {2,3}

#if USE_TDM
  if (wave == 0) {  // prologue: DMA first tile pair into buffer 0
    tdm_load_2d((unsigned)(uintptr_t)&lsA[0][0], Ab + (size_t)tileM * NMAT,
                16u, 64u, 3u /*every 16 DW*/, 0u /*+1 DW*/);
    tdm_load_2d((unsigned)(uintptr_t)&lsB[0][0], Bb + tileN,
                64u, 16u, 5u /*every 64 DW*/, 3u /*+4 DW*/);
  }
#endif

  for (int kt = 0; kt < NMAT / 16; ++kt) {
    const int k0 = kt * 16;
    const float* bufA = lsA[kt & 1];
    const float* bufB = lsB[kt & 1];

#if USE_TDM
    if (wave == 0) {
      if (kt + 1 < NMAT / 16) {  // DMA next tile pair into the other buffer
        tdm_load_2d((unsigned)(uintptr_t)&lsA[(kt + 1) & 1][0],
                    Ab + (size_t)tileM * NMAT + (k0 + 16), 16u, 64u, 3u, 0u);
        tdm_load_2d((unsigned)(uintptr_t)&lsB[(kt + 1) & 1][0],
                    Bb + (size_t)(k0 + 16) * NMAT + tileN, 64u, 16u, 5u, 3u);
        __builtin_amdgcn_s_wait_tensorcnt(2);  // current pair landed
      } else {
        __builtin_amdgcn_s_wait_tensorcnt(0);
      }
    }
    __syncthreads();
#else
    // fallback: direct staging (128 threads x 2 float4 per tile)
#pragma unroll
    for (int i = 0; i < 2; ++i) {
      const int f = tid + i * 128;            // 0..255
      {
        const int row = f >> 2;               // 0..63
        const int kq  = (f & 3) << 2;         // 0,4,8,12
        const float4 v = *reinterpret_cast<const float4*>(
            Ab + (size_t)(tileM + row) * NMAT + k0 + kq);
        float* dst = const_cast<float*>(&bufA[row * LDA_P + kq]);
        dst[0] = v.x; dst[1] = v.y; dst[2] = v.z; dst[3] = v.w;
      }
      {
        const int kr = f >> 4;                // 0..15
        const int cq = (f & 15) << 2;         // 0..60
        const float4 v = *reinterpret_cast<const float4*>(
            Bb + (size_t)(k0 + kr) * NMAT + tileN + cq);
        *reinterpret_cast<float4*>(const_cast<float*>(&bufB[kr * LDB_P + cq])) = v;
      }
    }
    if (k0 + 16 < NMAT) {  // global_prefetch_b8 of the next K tiles
      __builtin_prefetch(Ab + (size_t)(tileM + (tid >> 1)) * NMAT + k0 + 16, 0, 0);
      __builtin_prefetch(Bb + (size_t)(k0 + 16 + (tid & 15)) * NMAT + tileN, 0, 0);
    }
    __syncthreads();
#endif

    // ---- 16 WMMA_F32_16x16x4 per wave per K-step ----
#pragma unroll
    for (int kk = 0; kk < 16; kk += 4) {
      const int ka = kk + kb;
      v2f a0, a1, b0, b1;
      a0.x = bufA[(wm + mr) * LDA_P + ka];
      a0.y = bufA[(wm + mr) * LDA_P + ka + 1];
      a1.x = bufA[(wm + 16 + mr) * LDA_P + ka];
      a1.y = bufA[(wm + 16 + mr) * LDA_P + ka + 1];
      b0.x = bufB[ka * LDB_P + wn + mr];
      b0.y = bufB[(ka + 1) * LDB_P + wn + mr];
      b1.x = bufB[ka * LDB_P + wn + 16 + mr];
      b1.y = bufB[(ka + 1) * LDB_P + wn + 16 + mr];
      acc[0][0] = __builtin_amdgcn_wmma_f32_16x16x4_f32(false, a0, false, b0,
                                                        (short)0, acc[0][0], false, false);
      acc[0][1] = __builtin_amdgcn_wmma_f32_16x16x4_f32(false, a0, false, b1,
                                                        (short)0, acc[0][1], false, false);
      acc[1][0] = __builtin_amdgcn_wmma_f32_16x16x4_f32(false, a1, false, b0,
                                                        (short)0, acc[1][0], false, false);
      acc[1][1] = __builtin_amdgcn_wmma_f32_16x16x4_f32(false, a1, false, b1,
                                                        (short)0, acc[1][1], false, false);
    }
    __syncthreads();
  }

  // ---- epilogue: D = alpha*acc + delta*I ----
  const int rh = (lane >> 4) << 3;  // rows r (lanes 0-15) / r+8 (lanes 16-31)
  const int cn = lane & 15;
#pragma unroll
  for (int ti = 0; ti < 2; ++ti) {
#pragma unroll
    for (int tj = 0; tj < 2; ++tj) {
      const int colg = tileN + wn + tj * 16 + cn;
#pragma unroll
      for (int r = 0; r < 8; ++r) {
        const int rowg = tileM + wm + ti * 16 + rh + r;
        const float v = alpha * acc[ti][tj][r] + ((rowg == colg) ? delta : 0.0f);
        Db[(size_t)rowg * NMAT + colg] = v;
      }
    }
  }
}

// ---------------------------------------------------------------------------
// Elementwise helper kernels (all grid: (NN/256 [, batch]), block 256)
// ---------------------------------------------------------------------------
__global__ void mean_kernel(const float* __restrict__ in, float* __restrict__ mean)
{
  const int i = blockIdx.x * 256 + threadIdx.x;
  float s = 0.0f;
#pragma unroll
  for (int b = 0; b < BATCH; ++b) s += in[(size_t)b * NN + i];
  mean[i] = s * (1.0f / BATCH);
}

// new_weight (-> d_out tail) plus reconditioned symmetric Xr (-> ws)
__global__ void nw_kernel(const float* __restrict__ w, const float* __restrict__ mean,
                          float* __restrict__ nw, float* __restrict__ Xr)
{
  const int c = blockIdx.y;
  const int i = blockIdx.x * 256 + threadIdx.x;
  const int r = i >> 9, col = i & 511;
  const size_t base = (size_t)c * NN;
  const int it = (col << 9) | r;
  const float a  = 0.5f * w[base + i]  + 0.5f * mean[i];
  const float at = 0.5f * w[base + it] + 0.5f * mean[it];
  nw[base + i] = a;
  Xr[base + i] = (1.0f - PSI) * 0.5f * (a + at) + ((r == col) ? PSI : 0.0f);
}

__global__ void zero_scal_kernel(float* __restrict__ s, int n)
{
  if ((int)threadIdx.x < n) s[threadIdx.x] = 0.0f;
}

__global__ void frob_kernel(const float* __restrict__ X, float* __restrict__ s)
{
  __shared__ float red[256];
  const int z = blockIdx.y;
  const int i = blockIdx.x * 256 + threadIdx.x;
  const float v = X[(size_t)z * NN + i];
  red[threadIdx.x] = v * v;
  __syncthreads();
  for (int st = 128; st > 0; st >>= 1) {
    if ((int)threadIdx.x < st) red[threadIdx.x] += red[threadIdx.x + st];
    __syncthreads();
  }
  if (threadIdx.x == 0) atomicAdd(&s[z], red[0]);
}

// Y = X / ||X||_F,  Z = I       (s[z] holds sum of squares)
__global__ void init_ns_kernel(const float* __restrict__ X, const float* __restrict__ s,
                               float* __restrict__ Y, float* __restrict__ Z)
{
  const int z = blockIdx.y;
  const int i = blockIdx.x * 256 + threadIdx.x;
  const size_t o = (size_t)z * NN + i;
  const float inv = rsqrtf(s[z]);  // 1/norm
  Y[o] = X[o] * inv;
  const int r = i >> 9, c = i & 511;
  Z[o] = (r == c) ? 1.0f : 0.0f;
}

// O = X * norm^{+1/2}  (pospow=1, sqrt finalize)  or  X * norm^{-1/2} (sri)
__global__ void scale_pow_kernel(const float* __restrict__ X, const float* __restrict__ s,
                                 float* __restrict__ O, int pospow)
{
  const int z = blockIdx.y;
  const int i = blockIdx.x * 256 + threadIdx.x;
  const float f = sqrtf(sqrtf(s[z]));   // (sumsq)^(1/4) = sqrt(norm)
  const float m = pospow ? f : (1.0f / f);
  O[(size_t)z * NN + i] = X[(size_t)z * NN + i] * m;
}

__global__ void symrecond_kernel(const float* __restrict__ Q, float* __restrict__ M)
{
  const int z = blockIdx.y;
  const int i = blockIdx.x * 256 + threadIdx.x;
  const int r = i >> 9, c = i & 511;
  const size_t base = (size_t)z * NN;
  const float a = Q[base + i];
  const float b = Q[base + ((c << 9) | r)];
  M[base + i] = (1.0f - PSI) * 0.5f * (a + b) + ((r == c) ? PSI : 0.0f);
}

__global__ void subI_kernel(const float* __restrict__ M, float* __restrict__ R)
{
  const int z = blockIdx.y;
  const int i = blockIdx.x * 256 + threadIdx.x;
  const int r = i >> 9, c = i & 511;
  const size_t o = (size_t)z * NN + i;
  R[o] = M[o] - ((r == c) ? 1.0f : 0.0f);
}

__global__ void setI_kernel(float* __restrict__ S, float cval)
{
  const int z = blockIdx.y;
  const int i = blockIdx.x * 256 + threadIdx.x;
  const int r = i >> 9, c = i & 511;
  S[(size_t)z * NN + i] = (r == c) ? cval : 0.0f;
}

// ---------------------------------------------------------------------------
// Host orchestration (graph-capture safe: kernel launches on `stream` only)
// ---------------------------------------------------------------------------
extern "C" void kernel_launch(void* const* d_in, const int* in_sizes, int n_in,
                              void* d_out, int out_size, void* d_ws, size_t ws_size,
                              hipStream_t stream)
{
  const float* input  = (const float*)d_in[0];   // (B, N, N)
  const float* weight = (const float*)d_in[1];   // (C, N, N)
  float* out = (float*)d_out;                    // (B, C, N, N)
  float* nw  = out + (size_t)BATCH * CCH * NN;   // (C, N, N) second tuple output

  float* ws   = (float*)d_ws;
  float* mean = ws;                              // NN
  float* sri  = mean + NN;                       // C*NN, persists through stage 3
  float* scal = sri + (size_t)CCH * NN;          // 256 scalars
  float* big  = scal + 256;
  size_t bigFloats = ws_size / sizeof(float);
  const size_t fixed = (size_t)(big - ws);
  bigFloats = (bigFloats > fixed) ? (bigFloats - fixed) : 0;

  const dim3 eb(256), eg1(NN / 256);
  const dim3 gb(128);

  // ---- stage 1: batch mean + EMA weight update + recondition ----
  float* Xr = big;  // C*NN (reused later by the chunked log stage)
  mean_kernel<<<eg1, eb, 0, stream>>>(input, mean);
  nw_kernel<<<dim3(NN / 256, CCH), eb, 0, stream>>>(weight, mean, nw, Xr);

  // ---- stage 2: sri = Xr^{-1/2} via coupled Newton-Schulz ----
  {
    float* Y0 = Xr + (size_t)CCH * NN;
    float* Y1 = Y0 + (size_t)CCH * NN;
    float* Z0 = Y1 + (size_t)CCH * NN;
    float* Z1 = Z0 + (size_t)CCH * NN;
    float* T  = Z1 + (size_t)CCH * NN;
    const dim3 egc(NN / 256, CCH), gg(8, 8, CCH);
    zero_scal_kernel<<<1, 256, 0, stream>>>(scal, CCH);
    frob_kernel<<<dim3(1024, CCH), eb, 0, stream>>>(Xr, scal);
    init_ns_kernel<<<egc, eb, 0, stream>>>(Xr, scal, Y0, Z0);
    float *Yc = Y0, *Yn = Y1, *Zc = Z0, *Zn = Z1;
    for (int it = 0; it < NS_IT_W; ++it) {
      gemm512_wmma<<<gg, gb, 0, stream>>>(Zc, 0, Yc, 0, T, 0, 0, -0.5f, 1.5f);
      gemm512_wmma<<<gg, gb, 0, stream>>>(Yc, 0, T, 0, Yn, 0, 0, 1.0f, 0.0f);
      gemm512_wmma<<<gg, gb, 0, stream>>>(T, 0, Zc, 0, Zn, 0, 0, 1.0f, 0.0f);
      float* t;
      t = Yc; Yc = Yn; Yn = t;
      t = Zc; Zc = Zn; Zn = t;
    }
    // sri = Z / sqrt(norm)   (Z ~ (Xr/norm)^{-1/2})
    scale_pow_kernel<<<egc, eb, 0, stream>>>(Zc, scal, sri, 0);
  }

  // ---- stages 3+4: conjugation and matrix log, chunked over 128 matrices ----
  const int total = BATCH * CCH;
  int G = (int)(bigFloats / (size_t)(8 * (size_t)NN));
  if (G < 1) G = 1;
  if (G > total) G = total;

  float* P   = big;
  float* Q   = P   + (size_t)G * NN;   // conj, later reused as R = M - I
  float* M   = Q   + (size_t)G * NN;
  float* cY0 = M   + (size_t)G * NN;
  float* cY1 = cY0 + (size_t)G * NN;
  float* cZ0 = cY1 + (size_t)G * NN;
  float* cZ1 = cZ0 + (size_t)G * NN;
  float* cT  = cZ1 + (size_t)G * NN;

  for (int g0 = 0; g0 < total; g0 += G) {
    const int g = (total - g0 < G) ? (total - g0) : G;
    const dim3 cg(8, 8, g), egg(NN / 256, g), fgg(1024, g);

    // conj = sri_c @ input_b @ sri_c
    gemm512_wmma<<<cg, gb, 0, stream>>>(sri, 1, input, 2, P, 0, g0, 1.0f, 0.0f);
    gemm512_wmma<<<cg, gb, 0, stream>>>(P, 0, sri, 1, Q, 0, g0, 1.0f, 0.0f);
    symrecond_kernel<<<egg, eb, 0, stream>>>(Q, M);

    // M <- M^{1/2^K_SQRT} via repeated normalized Newton-Schulz sqrt
    for (int j = 0; j < K_SQRT; ++j) {
      zero_scal_kernel<<<1, 256, 0, stream>>>(scal, g);
      frob_kernel<<<fgg, eb, 0, stream>>>(M, scal);
      init_ns_kernel<<<egg, eb, 0, stream>>>(M, scal, cY0, cZ0);
      float *Yc = cY0, *Yn = cY1, *Zc = cZ0, *Zn = cZ1;
      for (int it = 0; it < NS_IT_L; ++it) {
        gemm512_wmma<<<cg, gb, 0, stream>>>(Zc, 0, Yc, 0, cT, 0, 0, -0.5f, 1.5f);
        gemm512_wmma<<<cg, gb, 0, stream>>>(Yc, 0, cT, 0, Yn, 0, 0, 1.0f, 0.0f);
        gemm512_wmma<<<cg, gb, 0, stream>>>(cT, 0, Zc, 0, Zn, 0, 0, 1.0f, 0.0f);
        float* t;
        t = Yc; Yc = Yn; Yn = t;
        t = Zc; Zc = Zn; Zn = t;
      }
      scale_pow_kernel<<<egg, eb, 0, stream>>>(Yc, scal, M, 1);  // M = sqrt(M)
    }

    // log(M_orig) ~ 2^K * log(I + R), Mercator series via Horner (GEMM/term)
    subI_kernel<<<egg, eb, 0, stream>>>(M, Q);                   // R in Q
    {
      const float cm = ((SER_M & 1) ? 1.0f : -1.0f) / (float)SER_M;
      setI_kernel<<<egg, eb, 0, stream>>>(cY0, cm);
      float *Sc = cY0, *Sn = cY1;
      for (int jj = SER_M - 1; jj >= 1; --jj) {
        const float cj = ((jj & 1) ? 1.0f : -1.0f) / (float)jj;
        gemm512_wmma<<<cg, gb, 0, stream>>>(Q, 0, Sc, 0, Sn, 0, 0, 1.0f, cj);
        float* t = Sc; Sc = Sn; Sn = t;
      }
      const float two_k = (float)(1 << K_SQRT);
      gemm512_wmma<<<cg, gb, 0, stream>>>(Q, 0, Sc, 0, out, 1, g0, two_k, 0.0f);
    }
  }
}